// MultiheadAttentionWeights_15539191677142
// MI455X (gfx1250) — compile-verified
//
#include <hip/hip_runtime.h>
#include <stdint.h>

typedef __attribute__((ext_vector_type(2))) float v2f;
typedef __attribute__((ext_vector_type(8))) float v8f;
typedef __attribute__((ext_vector_type(2))) int   v2i;

#define EMBED_DIM 512
#define QDIM      512
#define XP_N      1024
#define B_        4
#define H_        128
#define W_        128
#define NTOK      (H_ * W_)   // 16384
#define NBT       256
#define TOPK      96
#define NUM_HEADS 8
#define QHD       64
#define KTOT      160         // 64 own + 96 gathered

#if defined(__gfx1250__) && __has_builtin(__builtin_amdgcn_global_load_async_to_lds_b64)
#define HAVE_ASYNC_LDS 1
#else
#define HAVE_ASYNC_LDS 0
#endif

static __device__ __forceinline__ v8f wmma_f32(v2f a, v2f b, v8f c) {
  // D = A(16x4 f32) * B(4x16 f32) + C(16x16 f32)
  return __builtin_amdgcn_wmma_f32_16x16x4_f32(false, a, false, b, (short)0, c,
                                               false, false);
}

// Copy 8 bytes global -> LDS, async (CDNA5 GLOBAL_LOAD_ASYNC_TO_LDS_B64) if
// the builtin exists, else through VGPRs.
static __device__ __forceinline__ void async_copy_b64(const float* gsrc, v2f* ldst) {
#if HAVE_ASYNC_LDS
  __builtin_amdgcn_global_load_async_to_lds_b64((v2i*)gsrc, (v2i*)ldst, 0, 0);
#else
  *ldst = *(const v2f*)gsrc;
#endif
}

static __device__ __forceinline__ void async_join() {
#if HAVE_ASYNC_LDS
  __builtin_amdgcn_s_wait_asynccnt(0);
#endif
  __syncthreads();
}

// ---------------------------------------------------------------------------
// Pre-pass: interleave W_in (512x1024) into K-pair-major v2f layout so a WMMA
// B fragment is a single b64 load:  Wt[p][n] = { W[2p][n], W[2p+1][n] }.
// ---------------------------------------------------------------------------
__global__ __launch_bounds__(256) void winterleave_kernel(
    const float* __restrict__ Wm, v2f* __restrict__ Wt)
{
  const int i = blockIdx.x * 256 + threadIdx.x;  // 0 .. 256*1024-1
  const int p = i >> 10;
  const int n = i & 1023;
  v2f v;
  v.x = Wm[(2 * p) * XP_N + n];
  v.y = Wm[(2 * p + 1) * XP_N + n];
  Wt[(long)p * XP_N + n] = v;
}

// ---------------------------------------------------------------------------
// Kernel A: xp[M=65536][1024] = x[M][512] @ W_in[512][1024] + b_in
// One wave computes a 32x64 tile (8 C frags); 8 waves/block => 256x64 block.
// Per K-step (K=4): 2 A b64 + 4 B b64 loads feed 8 WMMAs.
// ---------------------------------------------------------------------------
__global__ __launch_bounds__(256) void xp_gemm_kernel(
    const float* __restrict__ x, const v2f* __restrict__ Wt,
    const float* __restrict__ bias, float* __restrict__ xp)
{
  const int wave = threadIdx.x >> 5;
  const int lane = threadIdx.x & 31;
  const int half = lane >> 4;
  const int l    = lane & 15;

  const long m0 = (long)blockIdx.x * 256 + wave * 32;
  const int  n0 = blockIdx.y * 64;

  const float* arow0 = x + (m0 + l) * (long)EMBED_DIM;
  const float* arow1 = arow0 + 16 * EMBED_DIM;
  const v2f*   wt    = Wt + n0 + l;

  v8f c00 = {}, c01 = {}, c02 = {}, c03 = {};
  v8f c10 = {}, c11 = {}, c12 = {}, c13 = {};

  for (int k = 0; k < EMBED_DIM; k += 4) {
    const int ka = k + half * 2;                 // this lane's K pair base
    const v2f a0 = *(const v2f*)(arow0 + ka);
    const v2f a1 = *(const v2f*)(arow1 + ka);
    const v2f* wp = wt + (long)((k >> 1) + half) * XP_N;
    const v2f b0 = wp[0];
    const v2f b1 = wp[16];
    const v2f b2 = wp[32];
    const v2f b3 = wp[48];
    c00 = wmma_f32(a0, b0, c00);
    c01 = wmma_f32(a0, b1, c01);
    c02 = wmma_f32(a0, b2, c02);
    c03 = wmma_f32(a0, b3, c03);
    c10 = wmma_f32(a1, b0, c10);
    c11 = wmma_f32(a1, b1, c11);
    c12 = wmma_f32(a1, b2, c12);
    c13 = wmma_f32(a1, b3, c13);
  }

  const float bv0 = bias[n0 + l];
  const float bv1 = bias[n0 + 16 + l];
  const float bv2 = bias[n0 + 32 + l];
  const float bv3 = bias[n0 + 48 + l];

#pragma unroll
  for (int r = 0; r < 8; ++r) {
    float* o0 = xp + (m0 + r + half * 8) * (long)XP_N + n0;
    float* o1 = o0 + 16 * XP_N;
    o0[l]      = c00[r] + bv0;
    o0[16 + l] = c01[r] + bv1;
    o0[32 + l] = c02[r] + bv2;
    o0[48 + l] = c03[r] + bv3;
    o1[l]      = c10[r] + bv0;
    o1[16 + l] = c11[r] + bv1;
    o1[32 + l] = c12[r] + bv2;
    o1[48 + l] = c13[r] + bv3;
  }
}

// ---------------------------------------------------------------------------
// Kernel B: per (head, batch, block): scores = q(64x64) @ k_full(64x160),
// softmax over 160. LDS tiles stored K-pair-major (v2f) so every WMMA A/B
// fragment is one ds_load_b64. q + own-block keys staged via async-to-LDS.
// ---------------------------------------------------------------------------
#define QSTR 34    // v2f per q row    (32 pairs + 2 pad)  -> 17,408 B
#define KSTR 162   // v2f per k row    (160 tokens + 2 pad)-> 41,472 B

__global__ __launch_bounds__(128) void attn_kernel(
    const float* __restrict__ xp, const int* __restrict__ indexes,
    const float* __restrict__ weights, float* __restrict__ out)
{
  __shared__ v2f q_lds[64 * QSTR];   // [token][dim pair]
  __shared__ v2f k_lds[32 * KSTR];   // [dim pair][token] (B-matrix layout)

  const int blk = blockIdx.x;   // 0..255
  const int b   = blockIdx.y;   // 0..3
  const int h   = blockIdx.z;   // 0..7
  const int tid = threadIdx.x;  // 0..127

  const int blkY = blk >> 4;
  const int blkX = blk & 15;
  const int qoff = h * QHD;
  const int koff = QDIM + h * QHD;

  // Stage own-block queries + keys (keys transposed) via async global->LDS.
  for (int i = tid; i < 64 * 32; i += 128) {
    const int t = i >> 5;          // token in 8x8 block
    const int p = i & 31;          // head-dim pair
    const int ty = t >> 3, tx = t & 7;
    const long row = (long)b * NTOK + (blkY * 8 + ty) * W_ + (blkX * 8 + tx);
    const float* xr = xp + row * (long)XP_N;
    async_copy_b64(xr + qoff + 2 * p, &q_lds[t * QSTR + p]);
    async_copy_b64(xr + koff + 2 * p, &k_lds[p * KSTR + t]);
  }
  // Gathered + weighted keys must pass through VGPRs (scale by weight).
  const int ibase = (b * NBT + blk) * TOPK;
  for (int i = tid; i < TOPK * 32; i += 128) {
    const int j = i >> 5;          // topk slot
    const int p = i & 31;          // head-dim pair
    const int tok = indexes[ibase + j];
    const float wg = weights[ibase + j];
    const float* src = xp + ((long)b * NTOK + tok) * (long)XP_N + koff + 2 * p;
    v2f v = *(const v2f*)src;
    v.x *= wg;
    v.y *= wg;
    k_lds[p * KSTR + 64 + j] = v;
  }
  async_join();

  const int wave = tid >> 5;
  const int lane = tid & 31;
  const int half = lane >> 4;
  const int l    = lane & 15;
  const int arow = wave * 16 + l;    // A-matrix row for this lane

  v8f acc[10];
#pragma unroll
  for (int nt = 0; nt < 10; ++nt) acc[nt] = (v8f){};

  for (int k = 0; k < QHD; k += 4) {
    const int p = (k >> 1) + half;             // this lane's K pair
    const v2f a = q_lds[arow * QSTR + p];
    const v2f* kp = &k_lds[p * KSTR + l];
#pragma unroll
    for (int nt = 0; nt < 10; ++nt)
      acc[nt] = wmma_f32(a, kp[nt * 16], acc[nt]);
  }

  // Row softmax: row r (VGPR index r) lives across one 16-lane half.
  const long obase = ((((long)h * B_ + b) * NBT + blk) * 64) * KTOT;
#pragma unroll
  for (int r = 0; r < 8; ++r) {
    float m = -3.402823466e38f;
#pragma unroll
    for (int nt = 0; nt < 10; ++nt) m = fmaxf(m, acc[nt][r]);
    m = fmaxf(m, __shfl_xor(m, 1, 32));
    m = fmaxf(m, __shfl_xor(m, 2, 32));
    m = fmaxf(m, __shfl_xor(m, 4, 32));
    m = fmaxf(m, __shfl_xor(m, 8, 32));

    float e[10];
    float s = 0.0f;
#pragma unroll
    for (int nt = 0; nt < 10; ++nt) {
      e[nt] = __expf(acc[nt][r] - m);
      s += e[nt];
    }
    s += __shfl_xor(s, 1, 32);
    s += __shfl_xor(s, 2, 32);
    s += __shfl_xor(s, 4, 32);
    s += __shfl_xor(s, 8, 32);
    const float inv = 1.0f / s;

    const int row = wave * 16 + half * 8 + r;
    float* orow = out + obase + (long)row * KTOT;
#pragma unroll
    for (int nt = 0; nt < 10; ++nt) orow[nt * 16 + l] = e[nt] * inv;
  }
}

// ---------------------------------------------------------------------------
extern "C" void kernel_launch(void* const* d_in, const int* in_sizes, int n_in,
                              void* d_out, int out_size, void* d_ws, size_t ws_size,
                              hipStream_t stream)
{
  const float* x   = (const float*)d_in[0];
  const int*   idx = (const int*)d_in[1];
  const float* wts = (const float*)d_in[2];
  const float* Wm  = (const float*)d_in[3];
  const float* bi  = (const float*)d_in[4];
  float* out = (float*)d_out;

  // ws layout: [ Wt : 2 MiB ][ xp : 256 MiB ]
  v2f*   Wt = (v2f*)d_ws;
  float* xp = (float*)((char*)d_ws + (size_t)EMBED_DIM * XP_N * sizeof(float));

  winterleave_kernel<<<1024, 256, 0, stream>>>(Wm, Wt);

  dim3 g1(256, 16);            // M tiles of 256, N tiles of 64
  xp_gemm_kernel<<<g1, 256, 0, stream>>>(x, Wt, bi, xp);

  dim3 g2(NBT, B_, NUM_HEADS); // one workgroup per (block, batch, head)
  attn_kernel<<<g2, 128, 0, stream>>>(xp, idx, wts, out);
}